// DiracScheduler_32495722561804
// MI455X (gfx1250) — compile-verified
//
#include <hip/hip_runtime.h>

// Problem constants (from setup_inputs)
#define N_EVENTS   64
#define START_SIZE 1024
#define N_SAMPLES  131072
#define FACTOR     (N_SAMPLES / START_SIZE)   // 128
#define ROW_F4     (N_SAMPLES / 4)            // 32768 float4 per row
#define BLOCK      256
#define F4T        4                          // float4 per thread in shift kernel

// ---------------------------------------------------------------------------
// Raw LDS byte offset for inline asm (async ops take LDS offsets, not flat
// pointers). C-style cast performs the generic->local addrspacecast.
// ---------------------------------------------------------------------------
typedef __attribute__((address_space(3))) void lds_void_t;

__device__ __forceinline__ unsigned lds_byte_addr(void* p) {
    return (unsigned)(unsigned long long)(lds_void_t*)p;
}

// ---------------------------------------------------------------------------
// Kernel 1: per-event first-occurrence argmax over pos[0,e,0:1024].
// argmax(softmax(x)) == argmax(x). delay[e] = argmax * FACTOR.
// One block per event, 256 threads, 4 elements each, LDS tree reduction.
// ---------------------------------------------------------------------------
__global__ __launch_bounds__(BLOCK)
void dirac_argmax_kernel(const float* __restrict__ pos, int* __restrict__ delay) {
    __shared__ float sv[BLOCK];
    __shared__ int   si[BLOCK];

    const float* row = pos + (size_t)blockIdx.x * START_SIZE;
    const int tid = threadIdx.x;

    float bestv = row[tid];
    int   besti = tid;
    #pragma unroll
    for (int i = tid + BLOCK; i < START_SIZE; i += BLOCK) {
        float v = row[i];
        if (v > bestv) { bestv = v; besti = i; }   // strict '>' => first occurrence
    }
    sv[tid] = bestv;
    si[tid] = besti;
    __syncthreads();

    for (int s = BLOCK / 2; s > 0; s >>= 1) {
        if (tid < s) {
            float ov = sv[tid + s];
            int   oi = si[tid + s];
            if (ov > sv[tid] || (ov == sv[tid] && oi < si[tid])) {
                sv[tid] = ov;
                si[tid] = oi;
            }
        }
        __syncthreads();
    }
    if (tid == 0) delay[blockIdx.x] = si[0] * FACTOR;
}

// ---------------------------------------------------------------------------
// Kernel 2: out[row, t] = (t >= d) ? events[row, t - d] : 0,   d = delay[e].
//
// d is a multiple of 128 floats (= 32 float4 = one wave32 of float4), so:
//   * every float4 is entirely "zero" or entirely "copy"
//   * each j-slice of a wave covers a contiguous 32-aligned t4 range ->
//     every branch below is wave-uniform; async ops run with full EXEC
//   * shifted source addresses stay 16B aligned
//
// Copy path uses the CDNA5 async data mover: 4 batched b128 async loads
// (global -> LDS, 2KB in flight per wave), one s_wait_asynccnt fence, then
// 4 batched b128 async stores (LDS -> global). Each lane owns private 16B
// LDS slots, so no barrier is needed. s_endpgm's implicit wait-idle drains
// the trailing async stores.
// ---------------------------------------------------------------------------
__global__ __launch_bounds__(BLOCK)
void dirac_shift_kernel(const float4* __restrict__ events,
                        const int*    __restrict__ delay,
                        float4*       __restrict__ out) {
    __shared__ __align__(16) float4 stage[F4T * BLOCK];   // 16 KB

    const int row  = blockIdx.y;                          // row = b*N_EVENTS + e
    const int e    = row & (N_EVENTS - 1);
    const int d4   = delay[e] >> 2;                       // multiple of 32 (wave-aligned)
    const int base = blockIdx.x * (BLOCK * F4T);

    const float4* src = events + (size_t)row * ROW_F4;
    float4*       dst = out    + (size_t)row * ROW_F4;

    const unsigned lds0 = lds_byte_addr(&stage[threadIdx.x]);

    // Phase 1: issue all async global->LDS loads for the copy region.
    #pragma unroll
    for (int j = 0; j < F4T; ++j) {
        const int t4 = base + j * BLOCK + threadIdx.x;
        if (t4 >= d4) {
            const float4* g   = src + (t4 - d4);
            const unsigned ld = lds0 + (unsigned)(j * BLOCK * sizeof(float4));
            asm volatile("global_load_async_to_lds_b128 %0, %1, off"
                         :: "v"(ld), "v"(g) : "memory");
        }
    }

    // Single fence: async LDS writes are unordered w.r.t. async LDS reads.
    asm volatile("s_wait_asynccnt 0" ::: "memory");

    // Phase 2: async LDS->global stores (copy) / plain zero stores (head).
    #pragma unroll
    for (int j = 0; j < F4T; ++j) {
        const int t4 = base + j * BLOCK + threadIdx.x;
        float4* o = dst + t4;
        if (t4 >= d4) {
            const unsigned ld = lds0 + (unsigned)(j * BLOCK * sizeof(float4));
            asm volatile("global_store_async_from_lds_b128 %0, %1, off"
                         :: "v"(o), "v"(ld) : "memory");
        } else {
            *o = make_float4(0.f, 0.f, 0.f, 0.f);
        }
    }
}

// ---------------------------------------------------------------------------
// Launch: inputs are {pos: [1,64,1024] f32, events: [B,64,131072] f32},
// output [B,64,131072] f32. Workspace holds delay[64] (int).
// ---------------------------------------------------------------------------
extern "C" void kernel_launch(void* const* d_in, const int* in_sizes, int n_in,
                              void* d_out, int out_size, void* d_ws, size_t ws_size,
                              hipStream_t stream) {
    const float* pos    = (const float*)d_in[0];
    const float* events = (const float*)d_in[1];
    float*       outp   = (float*)d_out;
    int*         delay  = (int*)d_ws;

    const int rows = in_sizes[1] / N_SAMPLES;             // B * N_EVENTS (= 128)

    dirac_argmax_kernel<<<dim3(N_EVENTS), dim3(BLOCK), 0, stream>>>(pos, delay);

    dim3 grid(ROW_F4 / (BLOCK * F4T), rows);              // (32, 128) blocks
    dirac_shift_kernel<<<grid, dim3(BLOCK), 0, stream>>>(
        (const float4*)events, delay, (float4*)outp);
}